// EdgeLevelGCN_51436528337343
// MI455X (gfx1250) — compile-verified
//
#include <hip/hip_runtime.h>

// EdgeLevelGCN on MI455X (gfx1250): A_t@X_t via fp32 WMMA 16x16x4, fused
// (agg@W_t + b_t) + ReLU. Types split across grid.y for machine fill; a tiny
// reduction kernel computes the mean over types from an 8MB workspace.

#define N_NODES 8192
#define T_NUM   4
#define D_ATT   64
#define KCHUNK  128
#define BLOCK_THREADS 64      // 2 waves (wave32)
#define ROWS_PER_WAVE 16
#define ROWS_PER_BLOCK 32
#define AST_STRIDE 66         // padded row stride for agg staging (bank-conflict avoidance)

typedef __attribute__((ext_vector_type(2))) float v2f;
typedef __attribute__((ext_vector_type(4))) float v4f;
typedef __attribute__((ext_vector_type(8))) float v8f;

__device__ __forceinline__ v8f wmma_f32_k4(v2f a, v2f b, v8f c) {
    // D = A(16x4,f32) * B(4x16,f32) + C(16x16,f32)
    return __builtin_amdgcn_wmma_f32_16x16x4_f32(
        false, a, false, b, (short)0, c, false, false);
}

// ---------------------------------------------------------------------------
// Per-type kernel: computes hpart[t] tile = 0.25 * relu(agg_t @ W_t + b_t)
// grid = (N/32, T), block = 64 threads (2 waves), one wave = 16x64 tile.
// ---------------------------------------------------------------------------
__global__ __launch_bounds__(BLOCK_THREADS)
void gcn_type_kernel(const float* __restrict__ features,  // [N, T, D]
                     const float* __restrict__ graphs,    // [T, N, N]
                     const float* __restrict__ Wt,        // [T, D, D]
                     const float* __restrict__ bias,      // [T, D]
                     float* __restrict__ hpart)           // [T, N, D]
{
    __shared__ float Bsw[KCHUNK * D_ATT];                    // 32 KB swizzled X chunk
    __shared__ float Ast[2][ROWS_PER_WAVE * AST_STRIDE];     // per-wave agg staging

    const int tid  = threadIdx.x;
    const int wave = tid >> 5;
    const int lane = tid & 31;
    const int lh   = lane >> 4;
    const int lr   = lane & 15;
    const int t    = blockIdx.y;
    const int row0 = blockIdx.x * ROWS_PER_BLOCK + wave * ROWS_PER_WAVE;

    const v8f zv = {};
    v8f acc[4];
#pragma unroll
    for (int nt = 0; nt < 4; ++nt) acc[nt] = zv;

    const float* __restrict__ gbase =
        graphs + (size_t)t * N_NODES * N_NODES + (size_t)(row0 + lr) * N_NODES + (lh * 2);

    for (int k0 = 0; k0 < N_NODES; k0 += KCHUNK) {
        __syncthreads();   // protect Bsw reuse across chunks

        // Cooperative swizzled fill: 128x64 f32, one float4 per thread per iter.
        // Fragment f = s*4+nt holds element (k = 4s + j + 2*(lane>>4),
        // n = 16nt + (lane&15)) at Bsw[f*64 + lane*2 + j].
#pragma unroll
        for (int it = 0; it < (KCHUNK * D_ATT) / (BLOCK_THREADS * 4); ++it) {
            const int linear = tid + it * BLOCK_THREADS;       // 0..2047
            const int krel   = linear >> 4;                    // 0..127
            const int n      = (linear & 15) << 2;             // 0..60, step 4
            const v4f v = *(const v4f*)(features
                            + (size_t)(k0 + krel) * (T_NUM * D_ATT)
                            + t * D_ATT + n);
            const int s  = krel >> 2;
            const int jr = krel & 3;
            const int j  = jr & 1;
            const int h  = jr >> 1;
            const int nt = n >> 4;
            const int lb = h * 16 + (n & 15);
            float* dst = &Bsw[(s * 4 + nt) * 64 + j];
            dst[(lb + 0) * 2] = v.x;
            dst[(lb + 1) * 2] = v.y;
            dst[(lb + 2) * 2] = v.z;
            dst[(lb + 3) * 2] = v.w;
        }
        __syncthreads();

        const float* __restrict__ ga = gbase + k0;
#pragma unroll 4
        for (int s = 0; s < KCHUNK / 4; ++s) {
            // graphs is a 1GB stream read once: non-temporal, keep L2 for X/W
            const v2f afrag = __builtin_nontemporal_load((const v2f*)(ga + s * 4));
            const float* bl = &Bsw[s * 256 + lane * 2];
#pragma unroll
            for (int nt = 0; nt < 4; ++nt) {
                const v2f bfrag = *(const v2f*)(bl + nt * 64); // conflict-free ds_load_b64
                acc[nt] = wmma_f32_k4(afrag, bfrag, acc[nt]);
            }
        }
    }

    // Stage agg tile (C layout: m = j + 8*lh, n = 16nt + lr) to padded LDS
    float* __restrict__ ast = Ast[wave];
#pragma unroll
    for (int nt = 0; nt < 4; ++nt)
#pragma unroll
        for (int j = 0; j < 8; ++j)
            ast[(j + 8 * lh) * AST_STRIDE + nt * 16 + lr] = acc[nt][j];
    // DS ops within a wave are in-order (DScnt): staging reads below are safe.

    // Second GEMM: z = agg(16x64) @ W_t(64x64), fp32 WMMA, K=64
    const float* __restrict__ w = Wt + t * (D_ATT * D_ATT);
    v8f z[4];
#pragma unroll
    for (int nt = 0; nt < 4; ++nt) z[nt] = zv;
#pragma unroll
    for (int s = 0; s < D_ATT / 4; ++s) {
        const int k = s * 4 + lh * 2;
        const v2f af = *(const v2f*)(ast + lr * AST_STRIDE + k);
#pragma unroll
        for (int nt = 0; nt < 4; ++nt) {
            v2f wf;
            wf.x = w[(k + 0) * D_ATT + nt * 16 + lr];   // VGPR0: K = 4s + 2*lh
            wf.y = w[(k + 1) * D_ATT + nt * 16 + lr];   // VGPR1: K = 4s + 2*lh + 1
            z[nt] = wmma_f32_k4(af, wf, z[nt]);
        }
    }

    // bias + ReLU, pre-scaled by 1/T, write per-type partial
    float* __restrict__ hp = hpart + (size_t)t * N_NODES * D_ATT;
#pragma unroll
    for (int nt = 0; nt < 4; ++nt) {
        const float bb = bias[t * D_ATT + nt * 16 + lr];
#pragma unroll
        for (int j = 0; j < 8; ++j) {
            const float hv = fmaxf(z[nt][j] + bb, 0.0f) * 0.25f;
            hp[(size_t)(row0 + j + 8 * lh) * D_ATT + nt * 16 + lr] = hv;
        }
    }
}

// ---------------------------------------------------------------------------
// Reduction: out = sum_t hpart[t]  (hpart already pre-scaled by 1/T)
// ---------------------------------------------------------------------------
__global__ __launch_bounds__(256)
void mean_kernel(const float* __restrict__ hpart, float* __restrict__ out)
{
    const size_t i = (size_t)blockIdx.x * 256 + threadIdx.x;   // float4 index
    const size_t stride = (size_t)N_NODES * D_ATT / 4;
    const v4f* p = (const v4f*)hpart;
    v4f a = p[i];
    a += p[i + stride];
    a += p[i + 2 * stride];
    a += p[i + 3 * stride];
    ((v4f*)out)[i] = a;
}

// ---------------------------------------------------------------------------
// Fallback: fully fused single kernel (used only if ws is too small).
// ---------------------------------------------------------------------------
__global__ __launch_bounds__(BLOCK_THREADS)
void gcn_fused_kernel(const float* __restrict__ features,
                      const float* __restrict__ graphs,
                      const float* __restrict__ Wt,
                      const float* __restrict__ bias,
                      float* __restrict__ out)
{
    __shared__ float Bsw[KCHUNK * D_ATT];
    __shared__ float Ast[2][ROWS_PER_WAVE * AST_STRIDE];

    const int tid  = threadIdx.x;
    const int wave = tid >> 5;
    const int lane = tid & 31;
    const int lh   = lane >> 4;
    const int lr   = lane & 15;
    const int row0 = blockIdx.x * ROWS_PER_BLOCK + wave * ROWS_PER_WAVE;

    const v8f zv = {};
    v8f hacc[4];
#pragma unroll
    for (int nt = 0; nt < 4; ++nt) hacc[nt] = zv;

    for (int t = 0; t < T_NUM; ++t) {
        v8f acc[4];
#pragma unroll
        for (int nt = 0; nt < 4; ++nt) acc[nt] = zv;

        const float* __restrict__ gbase =
            graphs + (size_t)t * N_NODES * N_NODES + (size_t)(row0 + lr) * N_NODES + (lh * 2);

        for (int k0 = 0; k0 < N_NODES; k0 += KCHUNK) {
            __syncthreads();
#pragma unroll
            for (int it = 0; it < (KCHUNK * D_ATT) / (BLOCK_THREADS * 4); ++it) {
                const int linear = tid + it * BLOCK_THREADS;
                const int krel   = linear >> 4;
                const int n      = (linear & 15) << 2;
                const v4f v = *(const v4f*)(features
                                + (size_t)(k0 + krel) * (T_NUM * D_ATT)
                                + t * D_ATT + n);
                const int s  = krel >> 2;
                const int jr = krel & 3;
                const int j  = jr & 1;
                const int h  = jr >> 1;
                const int nt = n >> 4;
                const int lb = h * 16 + (n & 15);
                float* dst = &Bsw[(s * 4 + nt) * 64 + j];
                dst[(lb + 0) * 2] = v.x;
                dst[(lb + 1) * 2] = v.y;
                dst[(lb + 2) * 2] = v.z;
                dst[(lb + 3) * 2] = v.w;
            }
            __syncthreads();

            const float* __restrict__ ga = gbase + k0;
#pragma unroll 4
            for (int s = 0; s < KCHUNK / 4; ++s) {
                const v2f afrag = __builtin_nontemporal_load((const v2f*)(ga + s * 4));
                const float* bl = &Bsw[s * 256 + lane * 2];
#pragma unroll
                for (int nt = 0; nt < 4; ++nt) {
                    const v2f bfrag = *(const v2f*)(bl + nt * 64);
                    acc[nt] = wmma_f32_k4(afrag, bfrag, acc[nt]);
                }
            }
        }

        float* __restrict__ ast = Ast[wave];
#pragma unroll
        for (int nt = 0; nt < 4; ++nt)
#pragma unroll
            for (int j = 0; j < 8; ++j)
                ast[(j + 8 * lh) * AST_STRIDE + nt * 16 + lr] = acc[nt][j];

        const float* __restrict__ w = Wt + t * (D_ATT * D_ATT);
        v8f z[4];
#pragma unroll
        for (int nt = 0; nt < 4; ++nt) z[nt] = zv;
#pragma unroll
        for (int s = 0; s < D_ATT / 4; ++s) {
            const int k = s * 4 + lh * 2;
            const v2f af = *(const v2f*)(ast + lr * AST_STRIDE + k);
#pragma unroll
            for (int nt = 0; nt < 4; ++nt) {
                v2f wf;
                wf.x = w[(k + 0) * D_ATT + nt * 16 + lr];
                wf.y = w[(k + 1) * D_ATT + nt * 16 + lr];
                z[nt] = wmma_f32_k4(af, wf, z[nt]);
            }
        }

#pragma unroll
        for (int nt = 0; nt < 4; ++nt) {
            const float bb = bias[t * D_ATT + nt * 16 + lr];
#pragma unroll
            for (int j = 0; j < 8; ++j)
                hacc[nt][j] += fmaxf(z[nt][j] + bb, 0.0f);
        }
    }

#pragma unroll
    for (int nt = 0; nt < 4; ++nt)
#pragma unroll
        for (int j = 0; j < 8; ++j)
            out[(size_t)(row0 + j + 8 * lh) * D_ATT + nt * 16 + lr] = hacc[nt][j] * 0.25f;
}

extern "C" void kernel_launch(void* const* d_in, const int* in_sizes, int n_in,
                              void* d_out, int out_size, void* d_ws, size_t ws_size,
                              hipStream_t stream) {
    (void)in_sizes; (void)n_in; (void)out_size;
    const float* features = (const float*)d_in[0];   // [N, T, D]
    const float* graphs   = (const float*)d_in[1];   // [T, N, N]
    const float* W        = (const float*)d_in[2];   // [T, D, D]
    const float* b        = (const float*)d_in[3];   // [T, D]
    float* out            = (float*)d_out;           // [N, D]

    const size_t need = (size_t)T_NUM * N_NODES * D_ATT * sizeof(float);  // 8 MB
    if (ws_size >= need && d_ws != nullptr) {
        // Type-split path: 1024 blocks (2048 waves) to fill all matrix pipes.
        dim3 grid(N_NODES / ROWS_PER_BLOCK, T_NUM);          // 256 x 4
        gcn_type_kernel<<<grid, BLOCK_THREADS, 0, stream>>>(
            features, graphs, W, b, (float*)d_ws);
        const int nvec = N_NODES * D_ATT / 4;                // 131072 float4
        mean_kernel<<<nvec / 256, 256, 0, stream>>>((const float*)d_ws, out);
    } else {
        gcn_fused_kernel<<<N_NODES / ROWS_PER_BLOCK, BLOCK_THREADS, 0, stream>>>(
            features, graphs, W, b, out);
    }
}